// VoxMorphLoss_90795608638033
// MI455X (gfx1250) — compile-verified
//
#include <hip/hip_runtime.h>
#include <stdint.h>

#define BATCH   16
#define HSZ     1024
#define WSZ     1024
#define OHSZ    1018
#define OWSZ    1018
#define TS      32          // output tile edge
#define ITS     40          // input tile edge = TS + 9 - 1
#define TPB     256
#define SMBLK   4096
#define NCCBLK  (BATCH * 32 * 32)
#define EPSF    2.220446049250313e-16f

// ---- CDNA5 async global->LDS copy (ASYNCcnt-tracked) ----
__device__ __forceinline__ void async_ld_f32(uint32_t lds_byte, const float* gptr) {
  // vdst = per-lane LDS byte offset, vaddr = per-lane 64-bit global address
  asm volatile("global_load_async_to_lds_b32 %0, %1, off"
               :: "v"(lds_byte), "v"(gptr) : "memory");
}

__device__ __forceinline__ void wait_async0() {
#if __has_builtin(__builtin_amdgcn_s_wait_asynccnt)
  __builtin_amdgcn_s_wait_asynccnt(0);
#else
  asm volatile("s_wait_asynccnt 0" ::: "memory");
#endif
}

// ---- Kernel 1: local NCC term over 9x9 box sums (separable, LDS-staged) ----
__global__ __launch_bounds__(TPB) void ncc_tile_kernel(
    const float* __restrict__ gI, const float* __restrict__ gJ,
    double* __restrict__ partial) {
  __shared__ float sI[ITS * ITS];
  __shared__ float sJ[ITS * ITS];
  __shared__ float hI [ITS * TS];
  __shared__ float hJ [ITS * TS];
  __shared__ float hII[ITS * TS];
  __shared__ float hJJ[ITS * TS];
  __shared__ float hIJ[ITS * TS];
  __shared__ double red[TPB];

  const int tid = threadIdx.x;
  const int ox0 = blockIdx.x * TS;
  const int oy0 = blockIdx.y * TS;
  const size_t img = (size_t)blockIdx.z * ((size_t)HSZ * WSZ);
  const float* I = gI + img;
  const float* J = gJ + img;

  // Uniform (scalar-branch) classification: does this tile touch the conv
  // zero-pad ring or run past the image edge?
  const bool interior = (oy0 >= 1) && (oy0 + ITS - 2 < HSZ) &&
                        (ox0 >= 1) && (ox0 + ITS - 2 < WSZ);

  if (interior) {
    // Fast path: full-EXEC async DMA, no zero-fill, no extra barrier.
    const float* Ib = I + (size_t)(oy0 - 1) * WSZ + (ox0 - 1);
    const float* Jb = J + (size_t)(oy0 - 1) * WSZ + (ox0 - 1);
    for (int e = tid; e < ITS * ITS; e += TPB) {
      const int r = e / ITS;
      const int c = e - r * ITS;
      const size_t off = (size_t)r * WSZ + c;
      async_ld_f32((uint32_t)(uintptr_t)(sI + e), Ib + off);
      async_ld_f32((uint32_t)(uintptr_t)(sJ + e), Jb + off);
    }
  } else {
    // Border path: zero-fill implements the conv zero padding (rows/cols -1
    // and 1024) and the out-of-image halo of edge tiles, then masked DMA.
    for (int e = tid; e < ITS * ITS; e += TPB) { sI[e] = 0.0f; sJ[e] = 0.0f; }
    __syncthreads();
    for (int e = tid; e < ITS * ITS; e += TPB) {
      const int r = e / ITS;
      const int c = e - r * ITS;
      const int gy = oy0 - 1 + r;
      const int gx = ox0 - 1 + c;
      if ((unsigned)gy < (unsigned)HSZ && (unsigned)gx < (unsigned)WSZ) {
        const size_t off = (size_t)gy * WSZ + gx;
        async_ld_f32((uint32_t)(uintptr_t)(sI + e), I + off);
        async_ld_f32((uint32_t)(uintptr_t)(sJ + e), J + off);
      }
    }
  }
  wait_async0();
  __syncthreads();

  // Horizontal 9-tap sums for I, J, I*I, J*J, I*J (40 rows x 32 output cols).
  for (int e = tid; e < ITS * TS; e += TPB) {
    const int r = e >> 5;
    const int c = e & (TS - 1);
    const float* pi = sI + r * ITS + c;
    const float* pj = sJ + r * ITS + c;
    float si = 0.f, sj = 0.f, sii = 0.f, sjj = 0.f, sij = 0.f;
#pragma unroll
    for (int t = 0; t < 9; ++t) {
      const float a = pi[t];
      const float b = pj[t];
      si += a; sj += b;
      sii = fmaf(a, a, sii);
      sjj = fmaf(b, b, sjj);
      sij = fmaf(a, b, sij);
    }
    hI[e] = si; hJ[e] = sj; hII[e] = sii; hJJ[e] = sjj; hIJ[e] = sij;
  }
  __syncthreads();

  // Vertical 9-tap sums + per-pixel NCC term (mirrors reference f32 math).
  double acc = 0.0;
  for (int o = tid; o < TS * TS; o += TPB) {
    const int r = o >> 5;
    const int c = o & (TS - 1);
    const int oy = oy0 + r;
    const int ox = ox0 + c;
    if (oy < OHSZ && ox < OWSZ) {
      float Is = 0.f, Js = 0.f, IIs = 0.f, JJs = 0.f, IJs = 0.f;
#pragma unroll
      for (int t = 0; t < 9; ++t) {
        const int idx = (r + t) * TS + c;
        Is  += hI [idx]; Js  += hJ [idx];
        IIs += hII[idx]; JJs += hJJ[idx]; IJs += hIJ[idx];
      }
      const float uI = Is / 81.0f;
      const float uJ = Js / 81.0f;
      const float cross = IJs - uJ * Is - uI * Js + uI * uJ * 81.0f;
      const float Ivar  = IIs - 2.0f * uI * Is + uI * uI * 81.0f;
      const float Jvar  = JJs - 2.0f * uJ * Js + uJ * uJ * 81.0f;
      const float term  = (cross * cross) / (Ivar * Jvar + EPSF);
      acc += (double)term;
    }
  }

  red[tid] = acc;
  __syncthreads();
  for (int s = TPB >> 1; s > 0; s >>= 1) {
    if (tid < s) red[tid] += red[tid + s];
    __syncthreads();
  }
  if (tid == 0) {
    partial[((size_t)blockIdx.z * gridDim.y + blockIdx.y) * gridDim.x + blockIdx.x] = red[0];
  }
}

// ---- Kernel 2: smoothness partial sums on y (served mostly from 192MB L2) ----
__global__ __launch_bounds__(TPB) void smooth_kernel(
    const float* __restrict__ y, double* __restrict__ partial2) {
  const size_t total  = (size_t)BATCH * HSZ * WSZ;
  const size_t stride = (size_t)gridDim.x * TPB;
  double ax = 0.0, ay = 0.0;
  for (size_t i = (size_t)blockIdx.x * TPB + threadIdx.x; i < total; i += stride) {
    __builtin_prefetch(y + i + stride, 0, 0);  // global_prefetch_b8 (speculative)
    const int c = (int)(i & (WSZ - 1));
    const int r = (int)((i >> 10) & (HSZ - 1));
    const float v = y[i];
    if (c < WSZ - 1) { const float d = y[i + 1]   - v; ax += (double)(d * d); }
    if (r < HSZ - 1) { const float d = y[i + WSZ] - v; ay += (double)(d * d); }
  }
  __shared__ double rx[TPB], ry[TPB];
  const int tid = threadIdx.x;
  rx[tid] = ax; ry[tid] = ay;
  __syncthreads();
  for (int s = TPB >> 1; s > 0; s >>= 1) {
    if (tid < s) { rx[tid] += rx[tid + s]; ry[tid] += ry[tid + s]; }
    __syncthreads();
  }
  if (tid == 0) {
    partial2[2 * blockIdx.x]     = rx[0];
    partial2[2 * blockIdx.x + 1] = ry[0];
  }
}

// ---- Kernel 3: deterministic final reduction -> scalar loss ----
__global__ __launch_bounds__(TPB) void finalize_kernel(
    const double* __restrict__ pcc, const double* __restrict__ psm,
    float* __restrict__ out) {
  __shared__ double r0[TPB], r1[TPB], r2[TPB];
  const int tid = threadIdx.x;
  double scc = 0.0, sdx = 0.0, sdy = 0.0;
  for (int i = tid; i < NCCBLK; i += TPB) scc += pcc[i];
  for (int i = tid; i < SMBLK;  i += TPB) { sdx += psm[2 * i]; sdy += psm[2 * i + 1]; }
  r0[tid] = scc; r1[tid] = sdx; r2[tid] = sdy;
  __syncthreads();
  for (int s = TPB >> 1; s > 0; s >>= 1) {
    if (tid < s) { r0[tid] += r0[tid + s]; r1[tid] += r1[tid + s]; r2[tid] += r2[tid + s]; }
    __syncthreads();
  }
  if (tid == 0) {
    const double cc  = r0[0] / ((double)BATCH * OHSZ * OWSZ);
    const double mdx = r1[0] / ((double)BATCH * HSZ * (WSZ - 1));
    const double mdy = r2[0] / ((double)BATCH * (HSZ - 1) * WSZ);
    const double sm  = 0.5 * (mdx + mdy);
    out[0] = (float)(-cc + 0.1 * sm);
  }
}

extern "C" void kernel_launch(void* const* d_in, const int* in_sizes, int n_in,
                              void* d_out, int out_size, void* d_ws, size_t ws_size,
                              hipStream_t stream) {
  (void)in_sizes; (void)n_in; (void)out_size; (void)ws_size;
  const float* y  = (const float*)d_in[0];
  const float* yt = (const float*)d_in[1];

  double* pcc = (double*)d_ws;          // NCCBLK doubles
  double* psm = pcc + NCCBLK;           // 2*SMBLK doubles

  dim3 grid1((OWSZ + TS - 1) / TS, (OHSZ + TS - 1) / TS, BATCH);  // 32 x 32 x 16
  ncc_tile_kernel<<<grid1, TPB, 0, stream>>>(y, yt, pcc);
  smooth_kernel<<<SMBLK, TPB, 0, stream>>>(y, psm);
  finalize_kernel<<<1, TPB, 0, stream>>>(pcc, psm, (float*)d_out);
}